// KDPointToPointLoss_39316130628223
// MI455X (gfx1250) — compile-verified
//
#include <hip/hip_runtime.h>

// KD point-to-point loss on gfx1250 (MI455X, wave32).
// d2 = |s|^2 + |t|^2 - 2 s.t ; loss[b] = mean_n min_m d2 / 3 -> (1/(3N)) sum_n min_m d2.
// Cross term computed with V_WMMA_F32_16X16X4_F32, folding |t|^2 into K=4.

typedef __attribute__((ext_vector_type(2))) float v2f;
typedef __attribute__((ext_vector_type(4))) float v4f;
typedef __attribute__((ext_vector_type(8))) float v8f;

#define BB 8
#define CC 3
#define NN 4096
#define MM 4096
#define SRC_TILES 2   // source 16-tiles per wave (reuses each A-tile load twice)

// Build augmented target buffer: aug[b*M+m] = {tx, ty, tz, |t|^2} (float4, 16B),
// and zero the 8-element output (harness poisons d_out).
__global__ void kd_aug_targets(const float* __restrict__ tgt,
                               float* __restrict__ aug,
                               float* __restrict__ out) {
  int i = blockIdx.x * blockDim.x + threadIdx.x;
  if (i < BB) out[i] = 0.0f;
  if (i >= BB * MM) return;
  int b = i >> 12;          // / 4096
  int m = i & (MM - 1);
  const float* p = tgt + (size_t)b * CC * MM + m;
  float x = p[0], y = p[MM], z = p[2 * MM];
  v4f o = {x, y, z, x * x + y * y + z * z};
  *(v4f*)(aug + (size_t)i * 4) = o;   // global_store_b128
}

__global__ void __launch_bounds__(256)
kd_nn_loss(const float* __restrict__ src,
           const float* __restrict__ aug,
           float* __restrict__ out) {
  const int lane = threadIdx.x & 31;
  const int wave = threadIdx.x >> 5;
  const int wid  = blockIdx.x * (blockDim.x >> 5) + wave;

  const int groups_per_batch = NN / (16 * SRC_TILES);   // 128
  const int b  = wid / groups_per_batch;
  const int n0 = (wid % groups_per_batch) * (16 * SRC_TILES);

  const int half = lane >> 4;   // 0 -> K=0,1 (and D rows M=r) ; 1 -> K=2,3 (rows M=r+8)
  const int l16  = lane & 15;

  const float* sb = src + (size_t)b * CC * NN;

  // B operands (4xK=16 source tile): B[k][n].  Layout: VGPR v, lanes 0-15 hold K=v,
  // lanes 16-31 hold K=v+2, column N = lane%16.
  // B row 0..2 = -2*s_xyz, row 3 = 1.0 (multiplies the |t|^2 slot of A).
  v2f bop[SRC_TILES];
#pragma unroll
  for (int t = 0; t < SRC_TILES; ++t) {
    int n = n0 + t * 16 + l16;
    float b0, b1;
    if (half == 0) {
      b0 = -2.0f * sb[0 * NN + n];   // K=0
      b1 = -2.0f * sb[1 * NN + n];   // K=1
    } else {
      b0 = -2.0f * sb[2 * NN + n];   // K=2
      b1 = 1.0f;                     // K=3
    }
    bop[t][0] = b0;
    bop[t][1] = b1;
  }

  const float* ab = aug + (size_t)b * MM * 4;

  float mn[SRC_TILES];
#pragma unroll
  for (int t = 0; t < SRC_TILES; ++t) mn[t] = __builtin_inff();

  const v8f czero = {};
  for (int m0 = 0; m0 < MM; m0 += 16) {
    // A operand (16 targets x K=4): lane<16 -> A[l16][0..1] = (tx,ty),
    // lane>=16 -> A[l16][2..3] = (tz,|t|^2).  One coalesced 64-bit load per lane.
    v2f a = *(const v2f*)(ab + (size_t)(m0 + l16) * 4 + (half ? 2 : 0));

#pragma unroll
    for (int t = 0; t < SRC_TILES; ++t) {
      v8f d = __builtin_amdgcn_wmma_f32_16x16x4_f32(
          false, a, false, bop[t], (short)0, czero, false, false);
      // D[m_row, n_col] = |t|^2 - 2 s.t ; lane holds rows r (+8 for upper half),
      // column n = n0 + t*16 + l16.  Running min over the 8 rows.
#pragma unroll
      for (int r = 0; r < 8; ++r) mn[t] = fminf(mn[t], d[r]);
    }
  }

  // Combine the two half-lane partials (rows r vs r+8): lanes l and l^16 cover the
  // same source column n.  After this, every lane holds the full min over M.
  float acc = 0.0f;
  const float inv3N = 1.0f / (float)(CC * NN);
#pragma unroll
  for (int t = 0; t < SRC_TILES; ++t) {
    float o = __shfl_xor(mn[t], 16, 32);
    float m = fminf(mn[t], o);
    int n = n0 + t * 16 + l16;
    float sx = sb[0 * NN + n], sy = sb[1 * NN + n], sz = sb[2 * NN + n];
    // d2_min = min(|t|^2 - 2 s.t) + |s|^2 ; equals |s - t_matched|^2 = per-point
    // summed squared diff, so no gather is needed.
    acc += (m + sx * sx + sy * sy + sz * sz) * inv3N;
  }

  // Width-16 reduction: lanes 0-15 and 16-31 hold identical duplicated data; the
  // lower group's lane 0 commits one atomic per wave.
#pragma unroll
  for (int k = 8; k >= 1; k >>= 1) acc += __shfl_xor(acc, k, 16);
  if (lane == 0) atomicAdd(&out[b], acc);
}

extern "C" void kernel_launch(void* const* d_in, const int* in_sizes, int n_in,
                              void* d_out, int out_size, void* d_ws, size_t ws_size,
                              hipStream_t stream) {
  const float* src = (const float*)d_in[0];   // [8,3,4096] f32
  const float* tgt = (const float*)d_in[1];   // [8,3,4096] f32
  float* out = (float*)d_out;                 // [8] f32
  float* aug = (float*)d_ws;                  // needs 8*4096*16 = 512 KB

  kd_aug_targets<<<(BB * MM + 255) / 256, 256, 0, stream>>>(tgt, aug, out);

  const int waves  = BB * (NN / (16 * SRC_TILES));  // 1024 waves
  const int blocks = waves / 8;                     // 256 threads = 8 waves/block
  kd_nn_loss<<<blocks, 256, 0, stream>>>(src, aug, out);
}